// PairwiseModel_16441134809766
// MI455X (gfx1250) — compile-verified
//
#include <hip/hip_runtime.h>
#include <stdint.h>

// ---------------------------------------------------------------------------
// Bilinear pairwise adjacency for MI455X (gfx1250), fp32 via V_WMMA_F32_16X16X4_F32
//   upper[p,q] = x_p^T Wu x_q + bu ;  lower[p,q] = x_p^T Wl x_q + bl
//   adj = triu(upper) + tril(lower,-1)
// Phase 0: Wt = W^T for both paths (one-shot, lets every LDS fragment be a
//          contiguous ds_load_b64 -- no VGPR packing movs in any inner loop)
// Phase 1: T{u,l} = X @ W{u,l}          (stored in d_ws)
// Phase 2: adj tile = T_rows @ X_cols^T (path picked by triangular position;
//          separate kernels for off-diagonal and diagonal 64x64 tiles)
// Staging: GLOBAL_LOAD_ASYNC_TO_LDS_B128 (ASYNCcnt) + double-buffered LDS so
// global latency overlaps the WMMA k-loop.
// ---------------------------------------------------------------------------

typedef float v2f __attribute__((ext_vector_type(2)));
typedef float v8f __attribute__((ext_vector_type(8)));

#define BN   16
#define NN   1024
#define DD   256
#define LDSS 68            // 64 + 4 pad: 16B-aligned rows, spreads LDS banks
#define TILE_F (64 * LDSS) // floats per staged 64x64 tile

__device__ __forceinline__ v8f wmma4(v2f a, v2f b, v8f c) {
  // D = A(16x4,f32) * B(4x16,f32) + C(16x16,f32)
  return __builtin_amdgcn_wmma_f32_16x16x4_f32(false, a, false, b, (short)0, c,
                                               false, false);
}

// Async-copy a 64x64 fp32 tile (source row stride DD floats) into LDS
// [64][LDSS]. 256 threads x 4 x 16B = 16 KB. Tracked by ASYNCcnt.
__device__ __forceinline__ void async_stage64x64(const float* src, float* dst,
                                                 int tid) {
  const int row = tid >> 2;
  const int q   = tid & 3;
#pragma unroll
  for (int j = 0; j < 4; ++j) {
    const int c = (q + 4 * j) << 2;                       // float col, mult of 4
    const uint32_t voff = (uint32_t)(row * (DD * 4) + c * 4);  // byte offset
    const uint32_t ldsa = (uint32_t)(uintptr_t)(dst + row * LDSS + c);
    // GVS mode: mem = SADDR(64b, uniform) + VADDR(32b) + IOFFSET
    asm volatile("global_load_async_to_lds_b128 %0, %1, %2 offset:0"
                 :: "v"(ldsa), "v"(voff), "s"(src)
                 : "memory");
  }
}

__device__ __forceinline__ void wait_async0() {
  asm volatile("s_wait_asynccnt 0x0" ::: "memory");
}

// 16x4 f32 fragment from row-major LDS tile lds[row][k]:
// lane<16: (m=lane, k0+0..1); lane>=16: (m=lane-16, k0+2..3) -> ds_load_b64.
// Used for A (rows of X or T) and for B via transposed storage (rows of X or
// Wt, since B[k][n] = rowsrc[n][k]).
__device__ __forceinline__ v2f frag_row(const float* lds, int rowBase, int k0,
                                        int lane) {
  const int r  = rowBase + (lane & 15);
  const int kk = k0 + ((lane >> 4) << 1);
  return *(const v2f*)(lds + r * LDSS + kk);
}

// ------------------- Phase 0: Wt[n][k] = W[k][n] (256x256) ------------------
__global__ __launch_bounds__(256) void transpose_w(
    const float* __restrict__ W, float* __restrict__ Wt) {
  const int k = blockIdx.x;   // source row
  const int n = threadIdx.x;  // source col
  Wt[(size_t)n * DD + k] = W[(size_t)k * DD + n];
}

// --------------------------- Phase 1: T = X @ Wt^T --------------------------
// grid: (DD/64=4, (B*N)/64=256); block 256 threads = 8 wave32.
// Wave w: m-tile = w&3, n-tiles = {2*(w>>2), 2*(w>>2)+1}
__global__ __launch_bounds__(256) void bilinear_phase1(
    const float* __restrict__ X, const float* __restrict__ Wt,
    float* __restrict__ T) {
  __shared__ float Xs[2][TILE_F];   // X rows  [m][k]
  __shared__ float Ws[2][TILE_F];   // Wt rows [n][k]

  const int tid  = threadIdx.x;
  const int lane = tid & 31;
  const int w    = tid >> 5;
  const int mt   = w & 3;
  const int nt0  = (w >> 2) << 1;
  const int row0 = blockIdx.y * 64;
  const int col0 = blockIdx.x * 64;

  v8f acc0 = {0.f, 0.f, 0.f, 0.f, 0.f, 0.f, 0.f, 0.f};
  v8f acc1 = acc0;

  const float* srcA = X + (size_t)row0 * DD;
  const float* srcB = Wt + (size_t)col0 * DD;  // rows n, cols k

  async_stage64x64(srcA, Xs[0], tid);
  async_stage64x64(srcB, Ws[0], tid);
  wait_async0();
  __syncthreads();

  for (int kc = 0; kc < DD / 64; ++kc) {
    if (kc + 1 < DD / 64) {  // prefetch next chunk into the other buffer
      async_stage64x64(srcA + (kc + 1) * 64, Xs[(kc + 1) & 1], tid);
      async_stage64x64(srcB + (kc + 1) * 64, Ws[(kc + 1) & 1], tid);
    }
    const float* xs = Xs[kc & 1];
    const float* ws = Ws[kc & 1];
#pragma unroll
    for (int ks = 0; ks < 16; ++ks) {
      const int k0 = ks << 2;
      const v2f a  = frag_row(xs, mt * 16, k0, lane);
      const v2f b0 = frag_row(ws, nt0 * 16, k0, lane);
      const v2f b1 = frag_row(ws, nt0 * 16 + 16, k0, lane);
      acc0 = wmma4(a, b0, acc0);
      acc1 = wmma4(a, b1, acc1);
    }
    if (kc + 1 < DD / 64) wait_async0();
    __syncthreads();
  }

  // C/D layout: VGPR v -> M = v + 8*(lane>=16), N = lane&15
#pragma unroll
  for (int v = 0; v < 8; ++v) {
    const int m = row0 + mt * 16 + v + ((lane >> 4) << 3);
    const int n = col0 + nt0 * 16 + (lane & 15);
    T[(size_t)m * DD + n]      = acc0[v];
    T[(size_t)m * DD + n + 16] = acc1[v];
  }
}

// -------------- Phase 2 (off-diagonal tiles): one path per tile -------------
// grid: (16,16,16) = (col tile C, row tile R, batch); early-out on R==C.
__global__ __launch_bounds__(256) void bilinear_phase2_offdiag(
    const float* __restrict__ Tu, const float* __restrict__ Tl,
    const float* __restrict__ X, const float* __restrict__ bUp,
    const float* __restrict__ bLo, float* __restrict__ out) {
  const int b = blockIdx.z, R = blockIdx.y, C = blockIdx.x;
  if (R == C) return;  // handled by the diagonal kernel

  __shared__ float As[2][TILE_F];  // T rows            [m][k]
  __shared__ float Bs[2][TILE_F];  // X rows (out cols) [n][k]

  const int tid  = threadIdx.x;
  const int lane = tid & 31;
  const int w    = tid >> 5;
  const int mt   = w & 3;
  const int nt0  = (w >> 2) << 1;

  const bool upper = (C > R);
  const float* Ta  = upper ? Tu : Tl;
  const float* srcA = Ta + ((size_t)b * NN + (size_t)R * 64) * DD;
  const float* srcB = X + ((size_t)b * NN + (size_t)C * 64) * DD;

  v8f acc0 = {0.f, 0.f, 0.f, 0.f, 0.f, 0.f, 0.f, 0.f};
  v8f acc1 = acc0;

  async_stage64x64(srcA, As[0], tid);
  async_stage64x64(srcB, Bs[0], tid);
  wait_async0();
  __syncthreads();

  for (int kc = 0; kc < DD / 64; ++kc) {
    if (kc + 1 < DD / 64) {
      async_stage64x64(srcA + (kc + 1) * 64, As[(kc + 1) & 1], tid);
      async_stage64x64(srcB + (kc + 1) * 64, Bs[(kc + 1) & 1], tid);
    }
    const float* as = As[kc & 1];
    const float* bs = Bs[kc & 1];
#pragma unroll
    for (int ks = 0; ks < 16; ++ks) {
      const int k0 = ks << 2;
      const v2f a  = frag_row(as, mt * 16, k0, lane);
      const v2f b0 = frag_row(bs, nt0 * 16, k0, lane);  // B = X^T fragment
      const v2f b1 = frag_row(bs, nt0 * 16 + 16, k0, lane);
      acc0 = wmma4(a, b0, acc0);
      acc1 = wmma4(a, b1, acc1);
    }
    if (kc + 1 < DD / 64) wait_async0();
    __syncthreads();
  }

  const float bias = upper ? bUp[0] : bLo[0];
  const int p0 = R * 64 + mt * 16;
  const int q0 = C * 64 + nt0 * 16;
  float* outb = out + (size_t)b * NN * NN;
#pragma unroll
  for (int v = 0; v < 8; ++v) {
    const int p = p0 + v + ((lane >> 4) << 3);
    const int q = q0 + (lane & 15);
    outb[(size_t)p * NN + q]      = acc0[v] + bias;
    outb[(size_t)p * NN + q + 16] = acc1[v] + bias;
  }
}

// ----------------- Phase 2 (diagonal tiles): both paths ---------------------
// grid: (16 diag tiles, 16 batches).
__global__ __launch_bounds__(256) void bilinear_phase2_diag(
    const float* __restrict__ Tu, const float* __restrict__ Tl,
    const float* __restrict__ X, const float* __restrict__ bUp,
    const float* __restrict__ bLo, float* __restrict__ out) {
  __shared__ float Au[2][TILE_F];  // Tu rows
  __shared__ float Al[2][TILE_F];  // Tl rows
  __shared__ float Bs[2][TILE_F];  // X rows

  const int tid  = threadIdx.x;
  const int lane = tid & 31;
  const int w    = tid >> 5;
  const int mt   = w & 3;
  const int nt0  = (w >> 2) << 1;
  const int t = blockIdx.x;  // R == C == t
  const int b = blockIdx.y;

  const float* srcU = Tu + ((size_t)b * NN + (size_t)t * 64) * DD;
  const float* srcL = Tl + ((size_t)b * NN + (size_t)t * 64) * DD;
  const float* srcB = X + ((size_t)b * NN + (size_t)t * 64) * DD;

  v8f acc0 = {0.f, 0.f, 0.f, 0.f, 0.f, 0.f, 0.f, 0.f};
  v8f acc1 = acc0, acc2 = acc0, acc3 = acc0;

  async_stage64x64(srcU, Au[0], tid);
  async_stage64x64(srcL, Al[0], tid);
  async_stage64x64(srcB, Bs[0], tid);
  wait_async0();
  __syncthreads();

  for (int kc = 0; kc < DD / 64; ++kc) {
    if (kc + 1 < DD / 64) {
      async_stage64x64(srcU + (kc + 1) * 64, Au[(kc + 1) & 1], tid);
      async_stage64x64(srcL + (kc + 1) * 64, Al[(kc + 1) & 1], tid);
      async_stage64x64(srcB + (kc + 1) * 64, Bs[(kc + 1) & 1], tid);
    }
    const float* au = Au[kc & 1];
    const float* al = Al[kc & 1];
    const float* bs = Bs[kc & 1];
#pragma unroll
    for (int ks = 0; ks < 16; ++ks) {
      const int k0 = ks << 2;
      const v2f a  = frag_row(au, mt * 16, k0, lane);
      const v2f a2 = frag_row(al, mt * 16, k0, lane);
      const v2f b0 = frag_row(bs, nt0 * 16, k0, lane);
      const v2f b1 = frag_row(bs, nt0 * 16 + 16, k0, lane);
      acc0 = wmma4(a, b0, acc0);
      acc1 = wmma4(a, b1, acc1);
      acc2 = wmma4(a2, b0, acc2);
      acc3 = wmma4(a2, b1, acc3);
    }
    if (kc + 1 < DD / 64) wait_async0();
    __syncthreads();
  }

  const float bu = bUp[0];
  const float bl = bLo[0];
  const int p0 = t * 64 + mt * 16;
  const int q0 = t * 64 + nt0 * 16;
  float* outb = out + (size_t)b * NN * NN;
#pragma unroll
  for (int v = 0; v < 8; ++v) {
    const int p = p0 + v + ((lane >> 4) << 3);
    const int q = q0 + (lane & 15);
    const float v0 = (q >= p) ? (acc0[v] + bu) : (acc2[v] + bl);
    const float v1 = (q + 16 >= p) ? (acc1[v] + bu) : (acc3[v] + bl);
    outb[(size_t)p * NN + q]      = v0;
    outb[(size_t)p * NN + q + 16] = v1;
  }
}

// ---------------------------------------------------------------------------
extern "C" void kernel_launch(void* const* d_in, const int* in_sizes, int n_in,
                              void* d_out, int out_size, void* d_ws,
                              size_t ws_size, hipStream_t stream) {
  // setup_inputs order: feats, W_upper, b_upper, W_lower, b_lower, length
  const float* feats = (const float*)d_in[0];
  const float* Wu    = (const float*)d_in[1];
  const float* bu    = (const float*)d_in[2];
  const float* Wl    = (const float*)d_in[3];
  const float* bl    = (const float*)d_in[4];
  float* out = (float*)d_out;

  // Workspace layout: Tu | Tl (16 MB each) | Wut | Wlt (256 KB each).
  // ws_size assumed >= 33 MB.
  float* Tu  = (float*)d_ws;
  float* Tl  = Tu + (size_t)BN * NN * DD;
  float* Wut = Tl + (size_t)BN * NN * DD;
  float* Wlt = Wut + (size_t)DD * DD;

  transpose_w<<<DD, DD, 0, stream>>>(Wu, Wut);
  transpose_w<<<DD, DD, 0, stream>>>(Wl, Wlt);

  dim3 g1(DD / 64, (BN * NN) / 64, 1);  // (4, 256)
  bilinear_phase1<<<g1, 256, 0, stream>>>(feats, Wut, Tu);
  bilinear_phase1<<<g1, 256, 0, stream>>>(feats, Wlt, Tl);

  dim3 g2(NN / 64, NN / 64, BN);        // (16, 16, 16), diag blocks early-out
  bilinear_phase2_offdiag<<<g2, 256, 0, stream>>>(Tu, Tl, feats, bu, bl, out);

  dim3 g2d(NN / 64, BN, 1);             // (16, 16) diagonal tiles
  bilinear_phase2_diag<<<g2d, 256, 0, stream>>>(Tu, Tl, feats, bu, bl, out);
}